// CGClass_49168785605355
// MI455X (gfx1250) — compile-verified
//
#include <hip/hip_runtime.h>
#include <stdint.h>

// ---------------------------------------------------------------------------
// SENSE-MRI CG reconstruction for MI455X (gfx1250).
// FFT2/IFFT2 as complex GEMMs with the 384-pt DFT matrix on the fp32 WMMA
// pipe (V_WMMA_F32_16X16X4_F32). GEMM is LDS-blocked (64x128 complex tile per
// 256-thread block, double-buffered async global->LDS staging) to cut L2
// operand re-reads ~5x; the problem is L2-traffic bound, math is free.
// ---------------------------------------------------------------------------

#define HN   384
#define HW   (HN * HN)        // 147456 = 576 * 256
#define NB   4
#define NC   16
#define NMAT (NB * NC)        // 64
#define CG_MAX_ITER 10
#define TOLF 1e-10f

// Set to 0 if the assembler rejects global_load_async_to_lds_b128.
#define USE_ASYNC_LDS 1

// ---- GEMM blocking ---------------------------------------------------------
#define BM  64                 // rows per block
#define BN  128                // cols per block
#define KB  32                 // K-block
#define A_LD 36                // LDS row stride (floats), bank-conflict-free
#define B_LD 144               // LDS row stride (floats), bank-conflict-free
#define A_PLANE (BM * A_LD)    // 2304 floats
#define B_PLANE (KB * B_LD)    // 4608 floats
#define BUF_FLOATS (2 * A_PLANE + 2 * B_PLANE)   // 13824 floats
#define SMEM_BYTES (2 * BUF_FLOATS * 4)          // 110592 B (double-buffered)

typedef float v2f __attribute__((ext_vector_type(2)));
typedef float v8f __attribute__((ext_vector_type(8)));

static __device__ __forceinline__ v8f wmma_f32_k4(v2f a, v2f b, v8f c) {
  return __builtin_amdgcn_wmma_f32_16x16x4_f32(
      /*neg_a=*/false, a, /*neg_b=*/false, b,
      /*c_mod=*/(short)0, c, /*reuse_a=*/false, /*reuse_b=*/false);
}

// Low 32 bits of a generic shared-aperture pointer == LDS byte offset
static __device__ __forceinline__ uint32_t lds_off(const float* p) {
  return (uint32_t)(uintptr_t)p;
}

// 16-byte global -> LDS copy. Async (ASYNCcnt-tracked) on CDNA5.
static __device__ __forceinline__ void cp16_g2s(float* ldsDst, const float* gSrc) {
#if USE_ASYNC_LDS
  asm volatile("global_load_async_to_lds_b128 %0, %1, off"
               :: "v"(lds_off(ldsDst)), "v"((uint64_t)(uintptr_t)gSrc)
               : "memory");
#else
  *(float4*)ldsDst = *(const float4*)gSrc;
#endif
}

static __device__ __forceinline__ void cp_wait_all() {
#if USE_ASYNC_LDS
  asm volatile("s_wait_asynccnt 0" ::: "memory");
#endif
}

// Stage one K-block (A: BM x KB, B: KB x BN, both complex) into an LDS buffer.
static __device__ __forceinline__ void stage_block(
    float* sbuf, const float* are, const float* aim,
    const float* bre, const float* bim,
    int m0, int n0, int k0, int tid) {
  float* As_re = sbuf;
  float* As_im = sbuf + A_PLANE;
  float* Bs_re = sbuf + 2 * A_PLANE;
  float* Bs_im = sbuf + 2 * A_PLANE + B_PLANE;
#pragma unroll
  for (int t = 0; t < (BM * KB / 4) / 256; ++t) {      // 2 float4s / thread
    const int chunk = tid + t * 256;
    const int row = chunk >> 3;                        // KB/4 = 8 chunks/row
    const int col = (chunk & 7) << 2;
    const int g = (m0 + row) * HN + k0 + col;
    cp16_g2s(&As_re[row * A_LD + col], &are[g]);
    cp16_g2s(&As_im[row * A_LD + col], &aim[g]);
  }
#pragma unroll
  for (int t = 0; t < (KB * BN / 4) / 256; ++t) {      // 4 float4s / thread
    const int chunk = tid + t * 256;
    const int row = chunk >> 5;                        // BN/4 = 32 chunks/row
    const int col = (chunk & 31) << 2;
    const int g = (k0 + row) * HN + n0 + col;
    cp16_g2s(&Bs_re[row * B_LD + col], &bre[g]);
    cp16_g2s(&Bs_im[row * B_LD + col], &bim[g]);
  }
}

// ---------------------------------------------------------------------------
// Build ortho DFT matrix F, its imag plane, and negated imag plane (conj F).
// ---------------------------------------------------------------------------
__global__ void k_build_dft(float* __restrict__ fre, float* __restrict__ fim,
                            float* __restrict__ fimn) {
  int idx = blockIdx.x * blockDim.x + threadIdx.x;
  if (idx >= HW) return;
  int j = idx / HN, k = idx % HN;
  int t = (j * k) % HN;
  float ang = -2.0f * 3.14159265358979323846f * (float)t / (float)HN;
  float s, c;
  sincosf(ang, &s, &c);
  const float scale = 0.05103103630798288f;  // 1/sqrt(384)
  fre[idx]  = c * scale;
  fim[idx]  = s * scale;
  fimn[idx] = -s * scale;
}

// ---------------------------------------------------------------------------
// Batched complex GEMM 384x384x384: C = A * B, fp32 WMMA, LDS double-buffered.
// aStride/bStride = 0 broadcasts the DFT matrix. Optional per-batch mask
// (batch = mat/NC) fused into the epilogue. Block: 256 thr, 64x128 out tile.
// Wave w: m-tile = w&3, four n-tiles at (w>>2)*64 (shared A fragment).
// ---------------------------------------------------------------------------
__global__ __launch_bounds__(256)
void k_zgemm384(const float* __restrict__ Are, const float* __restrict__ Aim,
                const float* __restrict__ Bre, const float* __restrict__ Bim,
                float* __restrict__ Cre, float* __restrict__ Cim,
                long long aStride, long long bStride,
                const float* __restrict__ maskBase) {
  extern __shared__ float smem[];
  const int tid  = threadIdx.x;
  const int wave = tid >> 5;
  const int lane = tid & 31;
  const int half = lane >> 4;
  const int r    = lane & 15;
  const int mat  = blockIdx.z;
  const int m0   = blockIdx.x * BM;
  const int n0   = blockIdx.y * BN;

  const int mi  = wave & 3;         // wave's m-tile within block (16 rows)
  const int nb0 = (wave >> 2) * 64; // wave's 4 n-tiles base col within block

  const float* are = Are + (long long)mat * aStride;
  const float* aim = Aim + (long long)mat * aStride;
  const float* bre = Bre + (long long)mat * bStride;
  const float* bim = Bim + (long long)mat * bStride;

  v8f zero = {0.f,0.f,0.f,0.f,0.f,0.f,0.f,0.f};
  v8f accRR[4], accII[4], accRI[4], accIR[4];
#pragma unroll
  for (int j = 0; j < 4; ++j) { accRR[j]=zero; accII[j]=zero; accRI[j]=zero; accIR[j]=zero; }

  stage_block(smem, are, aim, bre, bim, m0, n0, 0, tid);

  const int aRowBase = (mi * 16 + r) * A_LD + 2 * half;

  for (int kb = 0; kb < HN / KB; ++kb) {
    const float* cur = smem + (kb & 1) * BUF_FLOATS;
    cp_wait_all();            // this wave's staging of `cur` complete
    __syncthreads();          // everyone's staging complete, prev compute done
    if (kb + 1 < HN / KB)
      stage_block(smem + ((kb + 1) & 1) * BUF_FLOATS, are, aim, bre, bim,
                  m0, n0, (kb + 1) * KB, tid);

    const float* As_re = cur;
    const float* As_im = cur + A_PLANE;
    const float* Bs_re = cur + 2 * A_PLANE;
    const float* Bs_im = cur + 2 * A_PLANE + B_PLANE;

#pragma unroll 2
    for (int kk = 0; kk < KB; kk += 4) {
      v2f ar, ai;
      const int ao = aRowBase + kk;
      ar[0] = As_re[ao];     ar[1] = As_re[ao + 1];
      ai[0] = As_im[ao];     ai[1] = As_im[ao + 1];
      const int bbase = (kk + 2 * half) * B_LD + nb0 + r;
#pragma unroll
      for (int j = 0; j < 4; ++j) {
        v2f br, bi;
        const int bo = bbase + j * 16;
        br[0] = Bs_re[bo];   br[1] = Bs_re[bo + B_LD];
        bi[0] = Bs_im[bo];   bi[1] = Bs_im[bo + B_LD];
        accRR[j] = wmma_f32_k4(ar, br, accRR[j]);
        accII[j] = wmma_f32_k4(ai, bi, accII[j]);
        accRI[j] = wmma_f32_k4(ar, bi, accRI[j]);
        accIR[j] = wmma_f32_k4(ai, br, accIR[j]);
      }
    }
  }

  const float* mrow =
      maskBase ? (maskBase + (long long)(mat / NC) * (long long)HW) : nullptr;
  float* cre = Cre + (long long)mat * (long long)HW;
  float* cim = Cim + (long long)mat * (long long)HW;
#pragma unroll
  for (int j = 0; j < 4; ++j) {
    const int col = n0 + nb0 + j * 16 + r;
#pragma unroll
    for (int i = 0; i < 8; ++i) {
      const int row = m0 + mi * 16 + i + 8 * half;
      const long long o = (long long)row * HN + col;
      float vr = accRR[j][i] - accII[j][i];
      float vi = accRI[j][i] + accIR[j][i];
      if (mrow) { float mm = mrow[o]; vr *= mm; vi *= mm; }
      cre[o] = vr;
      cim[o] = vi;
    }
  }
}

// ---------------------------------------------------------------------------
// Pointwise kernels
// ---------------------------------------------------------------------------
__global__ void k_smul(const float* __restrict__ sre, const float* __restrict__ sim,
                       const float* __restrict__ pre, const float* __restrict__ pim,
                       float* __restrict__ tre, float* __restrict__ tim) {
  long long idx = (long long)blockIdx.x * blockDim.x + threadIdx.x;
  if (idx >= (long long)NMAT * HW) return;
  long long b   = idx / ((long long)NC * HW);
  long long pix = idx % HW;
  float pr = pre[b * HW + pix], pi = pim[b * HW + pix];
  float sr = sre[idx], si = sim[idx];
  tre[idx] = sr * pr - si * pi;
  tim[idx] = sr * pi + si * pr;
}

__global__ void k_masky(const float* __restrict__ yre, const float* __restrict__ yim,
                        const float* __restrict__ mask,
                        float* __restrict__ ore, float* __restrict__ oim) {
  long long idx = (long long)blockIdx.x * blockDim.x + threadIdx.x;
  if (idx >= (long long)NMAT * HW) return;
  long long b   = idx / ((long long)NC * HW);
  long long pix = idx % HW;
  float m = mask[b * HW + pix];
  ore[idx] = yre[idx] * m;
  oim[idx] = yim[idx] * m;
}

// LDS tree reduction + one fp32 global atomic per block.
__device__ __forceinline__ void block_reduce_add(float v, float* target) {
  __shared__ float sm[256];
  const int t = threadIdx.x;
  sm[t] = v;
  __syncthreads();
#pragma unroll
  for (int o = 128; o > 0; o >>= 1) {
    if (t < o) sm[t] += sm[t + o];
    __syncthreads();
  }
  if (t == 0) atomicAdd(target, sm[0]);
}

// rhs = AH(y) + lam*x_in; writes r = p = rhs, x = 0, reduces rTr = ||rhs||^2
__global__ __launch_bounds__(256)
void k_rhs_init(const float* __restrict__ gre, const float* __restrict__ gim,
                const float* __restrict__ sre, const float* __restrict__ sim,
                const float* __restrict__ xre_in, const float* __restrict__ xim_in,
                const float* __restrict__ lam,
                float* __restrict__ xre, float* __restrict__ xim,
                float* __restrict__ rre, float* __restrict__ rim,
                float* __restrict__ pre, float* __restrict__ pim,
                float* __restrict__ rTr) {
  const int b   = blockIdx.y;
  const int pix = blockIdx.x * 256 + threadIdx.x;  // HW == 576*256 exactly
  float accr = 0.f, acci = 0.f;
  const long long base = (long long)b * NC * HW + pix;
#pragma unroll
  for (int c = 0; c < NC; ++c) {
    const long long o = base + (long long)c * HW;
    float gr = gre[o], gi = gim[o], sr = sre[o], si = sim[o];
    accr += sr * gr + si * gi;   // conj(s) * g
    acci += sr * gi - si * gr;
  }
  const float l = lam[0];
  const long long po = (long long)b * HW + pix;
  accr += l * xre_in[po];
  acci += l * xim_in[po];
  rre[po] = accr; rim[po] = acci;
  pre[po] = accr; pim[po] = acci;
  xre[po] = 0.f;  xim[po] = 0.f;
  block_reduce_add(accr * accr + acci * acci, &rTr[b]);
}

// Mp = sum_c conj(s_c)*G_c + lam*p ; reduces pAp = Re<p, Mp>
__global__ __launch_bounds__(256)
void k_mp_dot(const float* __restrict__ gre, const float* __restrict__ gim,
              const float* __restrict__ sre, const float* __restrict__ sim,
              const float* __restrict__ pre, const float* __restrict__ pim,
              const float* __restrict__ lam,
              float* __restrict__ mre, float* __restrict__ mim,
              float* __restrict__ pAp) {
  const int b   = blockIdx.y;
  const int pix = blockIdx.x * 256 + threadIdx.x;
  float accr = 0.f, acci = 0.f;
  const long long base = (long long)b * NC * HW + pix;
#pragma unroll
  for (int c = 0; c < NC; ++c) {
    const long long o = base + (long long)c * HW;
    float gr = gre[o], gi = gim[o], sr = sre[o], si = sim[o];
    accr += sr * gr + si * gi;
    acci += sr * gi - si * gr;
  }
  const long long po = (long long)b * HW + pix;
  const float l  = lam[0];
  const float pr = pre[po], pi = pim[po];
  accr += l * pr;
  acci += l * pi;
  mre[po] = accr; mim[po] = acci;
  block_reduce_add(pr * accr + pi * acci, &pAp[b]);
}

// x += alpha*p; r -= alpha*Mp; reduce rTrNew (predicated on rTr > TOL)
__global__ __launch_bounds__(256)
void k_update_xr(float* __restrict__ xre, float* __restrict__ xim,
                 float* __restrict__ rre, float* __restrict__ rim,
                 const float* __restrict__ pre, const float* __restrict__ pim,
                 const float* __restrict__ mre, const float* __restrict__ mim,
                 const float* __restrict__ rTr, const float* __restrict__ pAp,
                 float* __restrict__ rTrNew) {
  const int b   = blockIdx.y;
  const int pix = blockIdx.x * 256 + threadIdx.x;
  const long long po = (long long)b * HW + pix;
  const float rt = rTr[b];          // uniform per block
  float contrib = 0.f;
  if (rt > TOLF) {
    const float alpha = rt / pAp[b];
    xre[po] += alpha * pre[po];
    xim[po] += alpha * pim[po];
    const float nr = rre[po] - alpha * mre[po];
    const float ni = rim[po] - alpha * mim[po];
    rre[po] = nr; rim[po] = ni;
    contrib = nr * nr + ni * ni;
  }
  block_reduce_add(contrib, &rTrNew[b]);
}

__global__ __launch_bounds__(256)
void k_update_p(float* __restrict__ pre, float* __restrict__ pim,
                const float* __restrict__ rre, const float* __restrict__ rim,
                const float* __restrict__ rTr, const float* __restrict__ rTrNew) {
  const int b   = blockIdx.y;
  const int pix = blockIdx.x * 256 + threadIdx.x;
  const float rt = rTr[b];
  if (rt <= TOLF) return;
  const float beta = rTrNew[b] / rt;
  const long long po = (long long)b * HW + pix;
  pre[po] = rre[po] + beta * pre[po];
  pim[po] = rim[po] + beta * pim[po];
}

__global__ void k_commit(float* __restrict__ rTr, const float* __restrict__ rTrNew) {
  const int b = threadIdx.x;
  if (b < NB && rTr[b] > TOLF) rTr[b] = rTrNew[b];
}

__global__ void k_zero(float* __restrict__ p, int n) {
  int i = blockIdx.x * blockDim.x + threadIdx.x;
  if (i < n) p[i] = 0.f;
}

__global__ void k_pack(const float* __restrict__ xre, const float* __restrict__ xim,
                       float* __restrict__ out) {
  long long idx = (long long)blockIdx.x * blockDim.x + threadIdx.x;
  if (idx >= (long long)NB * HW) return;
  out[2 * idx]     = xre[idx];
  out[2 * idx + 1] = xim[idx];
}

// ---------------------------------------------------------------------------
// Host-side orchestration (all launches on `stream`, graph-capturable).
// ---------------------------------------------------------------------------
extern "C" void kernel_launch(void* const* d_in, const int* in_sizes, int n_in,
                              void* d_out, int out_size, void* d_ws, size_t ws_size,
                              hipStream_t stream) {
  const float* lam   = (const float*)d_in[0];
  const float* xre_i = (const float*)d_in[1];
  const float* xim_i = (const float*)d_in[2];
  const float* yre   = (const float*)d_in[3];
  const float* yim   = (const float*)d_in[4];
  const float* sre   = (const float*)d_in[5];
  const float* sim   = (const float*)d_in[6];
  const float* mask  = (const float*)d_in[7];

  float* w = (float*)d_ws;
  float* Fre  = w; w += HW;
  float* Fim  = w; w += HW;
  float* Fimn = w; w += HW;                       // conj(F) imag plane
  float* b1re = w; w += (long long)NMAT * HW;     // ping-pong buffers [B,C,H,W]
  float* b1im = w; w += (long long)NMAT * HW;
  float* b2re = w; w += (long long)NMAT * HW;
  float* b2im = w; w += (long long)NMAT * HW;
  float* xre  = w; w += (long long)NB * HW;       // CG state [B,H,W]
  float* xim  = w; w += (long long)NB * HW;
  float* rre  = w; w += (long long)NB * HW;
  float* rim  = w; w += (long long)NB * HW;
  float* pre  = w; w += (long long)NB * HW;
  float* pim  = w; w += (long long)NB * HW;
  float* mre  = w; w += (long long)NB * HW;
  float* mim  = w; w += (long long)NB * HW;
  float* rTr  = w; w += NB;                       // scalars (contiguous)
  float* pAp  = w; w += NB;
  float* rNew = w; w += NB;

  const dim3 gG(HN / BM, HN / BN, NMAT), bG(256);         // GEMM grid 6x3x64
  const dim3 gP((unsigned)(((long long)NMAT * HW + 255) / 256)), bP(256);
  const dim3 gR(HW / 256, NB), bR(256);

  k_build_dft<<<(HW + 255) / 256, 256, 0, stream>>>(Fre, Fim, Fimn);

  // rhs = AH(y) + lam*x :  conj(F) * (mask .* y) * conj(F), coil-combine
  k_masky<<<gP, bP, 0, stream>>>(yre, yim, mask, b1re, b1im);
  k_zgemm384<<<gG, bG, SMEM_BYTES, stream>>>(Fre, Fimn, b1re, b1im, b2re, b2im,
                                             0, (long long)HW, nullptr);
  k_zgemm384<<<gG, bG, SMEM_BYTES, stream>>>(b2re, b2im, Fre, Fimn, b1re, b1im,
                                             (long long)HW, 0, nullptr);
  k_zero<<<1, 32, 0, stream>>>(rTr, 3 * NB);
  k_rhs_init<<<gR, bR, 0, stream>>>(b1re, b1im, sre, sim, xre_i, xim_i, lam,
                                    xre, xim, rre, rim, pre, pim, rTr);

  for (int it = 0; it < CG_MAX_ITER; ++it) {
    k_zero<<<1, 32, 0, stream>>>(pAp, 2 * NB);  // zero pAp + rNew
    // G = conj(F) * (mask .* (F * (s.*p) * F)) * conj(F)   (mask^2 == mask)
    k_smul<<<gP, bP, 0, stream>>>(sre, sim, pre, pim, b1re, b1im);
    k_zgemm384<<<gG, bG, SMEM_BYTES, stream>>>(Fre, Fim, b1re, b1im, b2re, b2im,
                                               0, (long long)HW, nullptr);
    k_zgemm384<<<gG, bG, SMEM_BYTES, stream>>>(b2re, b2im, Fre, Fim, b1re, b1im,
                                               (long long)HW, 0, mask);
    k_zgemm384<<<gG, bG, SMEM_BYTES, stream>>>(Fre, Fimn, b1re, b1im, b2re, b2im,
                                               0, (long long)HW, nullptr);
    k_zgemm384<<<gG, bG, SMEM_BYTES, stream>>>(b2re, b2im, Fre, Fimn, b1re, b1im,
                                               (long long)HW, 0, nullptr);
    k_mp_dot<<<gR, bR, 0, stream>>>(b1re, b1im, sre, sim, pre, pim, lam,
                                    mre, mim, pAp);
    k_update_xr<<<gR, bR, 0, stream>>>(xre, xim, rre, rim, pre, pim, mre, mim,
                                       rTr, pAp, rNew);
    k_update_p<<<gR, bR, 0, stream>>>(pre, pim, rre, rim, rTr, rNew);
    k_commit<<<1, NB, 0, stream>>>(rTr, rNew);
  }

  k_pack<<<((unsigned)(((long long)NB * HW + 255) / 256)), 256, 0, stream>>>(
      xre, xim, (float*)d_out);
}